// RBFEmbedding_68917045231809
// MI455X (gfx1250) — compile-verified
//
#include <hip/hip_runtime.h>

// RBF embedding fused kernel for gfx1250 (MI455X).
// Phase 1: per-row RBF mean (exp2-based Gaussian expansion, 50 reg accumulators/lane,
//          wave32 butterfly reduction) -> LDS tile [16 x 52(+pad)].
// Phase 2: projection GEMM via V_WMMA_F32_16X16X4_F32, C preloaded with
//          bias + token embedding so the WMMA result is the final output.

#define NN        512
#define D_MODEL   256
#define NUM_RBF   50
#define CUTOFF    12.0f

typedef __attribute__((ext_vector_type(2))) float v2f;
typedef __attribute__((ext_vector_type(8))) float v8f;

__global__ __launch_bounds__(256) void rbf_embed_kernel(
    const int*   __restrict__ atomic_numbers,  // [B, N]
    const float* __restrict__ positions,       // [B, N, 3]
    const float* __restrict__ token_emb,       // [100, 256]
    const float* __restrict__ proj_w,          // [50, 256]
    const float* __restrict__ proj_b,          // [256]
    float*       __restrict__ out)             // [B, N, 256]
{
    constexpr int TILE_M = 16;
    constexpr int RBF_STRIDE = 68;   // floats; conflict-free for WMMA A reads

    __shared__ float s_px[NN], s_py[NN], s_pz[NN];
    __shared__ float s_rbf[TILE_M * RBF_STRIDE];

    const int tid  = threadIdx.x;
    const int wave = tid >> 5;
    const int lane = tid & 31;

    const int tiles_per_batch = NN / TILE_M;               // 32
    const int b      = blockIdx.x / tiles_per_batch;
    const int i_base = (blockIdx.x % tiles_per_batch) * TILE_M;

    // ---- stage batch positions into LDS (SoA) ----
    const float* pb = positions + (size_t)b * NN * 3;
    for (int j = tid; j < NN; j += 256) {
        s_px[j] = pb[3 * j + 0];
        s_py[j] = pb[3 * j + 1];
        s_pz[j] = pb[3 * j + 2];
    }
    for (int idx = tid; idx < TILE_M * RBF_STRIDE; idx += 256)
        s_rbf[idx] = 0.0f;   // zero incl. K padding (k = 50..)
    __syncthreads();

    // ---- phase 1: RBF expansion + neighbor mean ----
    const float step  = CUTOFF / (float)(NUM_RBF - 1);
    const float width = step * 2.0f;
    const float nq    = -1.44269504088896f / (width * width);  // -log2(e)/w^2

    for (int rr = 0; rr < 2; ++rr) {
        const int row = wave * 2 + rr;            // 0..15 within tile
        const int i   = i_base + row;
        const float xi = s_px[i], yi = s_py[i], zi = s_pz[i];

        float acc[NUM_RBF];
        #pragma unroll
        for (int k = 0; k < NUM_RBF; ++k) acc[k] = 0.0f;

        for (int jj = 0; jj < NN / 32; ++jj) {
            const int j = lane + jj * 32;
            const float dx = xi - s_px[j];
            const float dy = yi - s_py[j];
            const float dz = zi - s_pz[j];
            const float d  = __builtin_amdgcn_sqrtf(
                __builtin_fmaf(dx, dx,
                  __builtin_fmaf(dy, dy,
                    __builtin_fmaf(dz, dz, 1e-8f))));
            #pragma unroll
            for (int k = 0; k < NUM_RBF; ++k) {
                const float t = d - (float)k * step;
                acc[k] += __builtin_amdgcn_exp2f(t * t * nq);   // v_exp_f32
            }
        }
        // wave32 butterfly reduction, lane 0 writes the mean
        #pragma unroll
        for (int k = 0; k < NUM_RBF; ++k) {
            float v = acc[k];
            #pragma unroll
            for (int off = 16; off > 0; off >>= 1)
                v += __shfl_xor(v, off, 32);
            if (lane == 0)
                s_rbf[row * RBF_STRIDE + k] = v * (1.0f / (float)NN);
        }
    }
    __syncthreads();

    // ---- phase 2: [16 x 52] @ [52 x 256] via V_WMMA_F32_16X16X4_F32 ----
    const int* an_row = atomic_numbers + (size_t)b * NN + i_base;
    const int  half   = lane >> 4;      // 0: lanes 0-15, 1: lanes 16-31
    const int  l16    = lane & 15;

    for (int t = 0; t < 2; ++t) {
        const int nb  = wave * 16 + t * 128;   // N-tile base (covers all 16 tiles)
        const int col = nb + l16;

        // C = bias + token embedding (per C/D layout: vgpr r -> M = r + 8*half)
        v8f c;
        #pragma unroll
        for (int r = 0; r < 8; ++r) {
            const int m   = r + half * 8;
            const int tok = an_row[m];
            c[r] = proj_b[col] + token_emb[(size_t)tok * D_MODEL + col];
        }

        // K loop: 13 steps of K=4 cover k = 0..51 (A LDS zero-padded, B guarded)
        #pragma unroll
        for (int kk = 0; kk < 13; ++kk) {
            const int ka = kk * 4 + half * 2;   // this lane's K pair
            v2f a;
            a.x = s_rbf[l16 * RBF_STRIDE + ka];
            a.y = s_rbf[l16 * RBF_STRIDE + ka + 1];
            v2f bf;
            bf.x = (ka     < NUM_RBF) ? proj_w[(size_t)(ka    ) * D_MODEL + col] : 0.0f;
            bf.y = (ka + 1 < NUM_RBF) ? proj_w[(size_t)(ka + 1) * D_MODEL + col] : 0.0f;
            c = __builtin_amdgcn_wmma_f32_16x16x4_f32(
                    false, a, false, bf, (short)0, c, false, false);
        }

        // store D (final output)
        #pragma unroll
        for (int r = 0; r < 8; ++r) {
            const int m = r + half * 8;
            const size_t orow = ((size_t)b * NN + (size_t)(i_base + m)) * D_MODEL;
            out[orow + col] = c[r];
        }
    }
}

extern "C" void kernel_launch(void* const* d_in, const int* in_sizes, int n_in,
                              void* d_out, int out_size, void* d_ws, size_t ws_size,
                              hipStream_t stream) {
    const int*   an   = (const int*)d_in[0];     // atomic_numbers [B,N] int32
    const float* pos  = (const float*)d_in[1];   // positions [B,N,3]
    const float* temb = (const float*)d_in[2];   // token_emb [100,256]
    const float* pw   = (const float*)d_in[3];   // proj_w [50,256]
    const float* pbv  = (const float*)d_in[4];   // proj_b [256]
    float* outp = (float*)d_out;

    const int B    = in_sizes[0] / NN;           // 16
    const int grid = B * (NN / 16);              // 512 blocks, 256 thr (8 waves)
    rbf_embed_kernel<<<grid, 256, 0, stream>>>(an, pos, temb, pw, pbv, outp);
}